// SuperPointMatching_53455162966422
// MI455X (gfx1250) — compile-verified
//
#include <hip/hip_runtime.h>
#include <cstdint>

#define RATIO_THRESHOLD 0.8f
#define EPS 1e-8f

#define BM 32      // ref rows per block (exclusive ownership -> no row atomics)
#define BN 64      // src cols per tile
#define KD 256     // feature dim (D)
#define LDA 260    // KD + 4 pad: 16B-aligned rows, bank-staggered
#define LDC 68     // C tile pad

typedef float v2f __attribute__((ext_vector_type(2)));
typedef float v8f __attribute__((ext_vector_type(8)));

#if __has_builtin(__builtin_amdgcn_global_load_async_to_lds_b128) && \
    __has_builtin(__builtin_amdgcn_s_wait_asynccnt)
#define SPM_ASYNC_LDS 1
typedef int spm_v4i __attribute__((vector_size(16)));
#endif

__device__ __forceinline__ unsigned f2ord(float f) {
    unsigned u = __float_as_uint(f);
    return (u & 0x80000000u) ? ~u : (u | 0x80000000u);
}

// Stage one float4 (16B) from global to LDS: async DMA if available, else copy.
__device__ __forceinline__ void stage16(const float* __restrict__ g, float* l) {
#ifdef SPM_ASYNC_LDS
    __builtin_amdgcn_global_load_async_to_lds_b128(
        (spm_v4i*)g, (spm_v4i*)l, 0, 0);
#else
    *(float4*)l = *(const float4*)g;
#endif
}

__device__ __forceinline__ void stage_wait() {
#ifdef SPM_ASYNC_LDS
    __builtin_amdgcn_s_wait_asynccnt(0);
#endif
}

// ---------------------------------------------------------------------------
// init: zero the per-column argmax table (atomicMax accumulator)
// ---------------------------------------------------------------------------
__global__ void spm_init_kernel(unsigned long long* colBest, int N) {
    int i = blockIdx.x * 256 + threadIdx.x;
    if (i < N) colBest[i] = 0ull;
}

// ---------------------------------------------------------------------------
// fused GEMM (f32 WMMA) + per-row top2 + per-column argmax
// ---------------------------------------------------------------------------
__global__ __launch_bounds__(256) void spm_gemm_kernel(
    const float* __restrict__ refF, const float* __restrict__ srcF,
    const unsigned char* __restrict__ refMask, const unsigned char* __restrict__ srcMask,
    int M, int N,
    unsigned long long* __restrict__ colBest,
    float* __restrict__ rowV0, float* __restrict__ rowV1, int* __restrict__ rowI0)
{
    __shared__ float As[BM * LDA];
    __shared__ float Bs[BN * LDA];
    __shared__ float Cs[BM * LDC];
    __shared__ float redV[BM * 8 * 2];
    __shared__ int   redI[BM * 8 * 2];

    const int t = threadIdx.x;
    const int rowBase = blockIdx.x * BM;

    // ---- prologue: stage A panel + first B panel (async if available) ----
    for (int q = t; q < BM * (KD / 4); q += 256) {
        int row = q / (KD / 4), c4 = q % (KD / 4);
        stage16(refF + (size_t)(rowBase + row) * KD + c4 * 4, &As[row * LDA + c4 * 4]);
    }
    for (int q = t; q < BN * (KD / 4); q += 256) {
        int row = q / (KD / 4), c4 = q % (KD / 4);
        stage16(srcF + (size_t)row * KD + c4 * 4, &Bs[row * LDA + c4 * 4]);
    }

    // per-thread packed masks (8 bytes each), loaded once / per tile as registers
    const int rowR = t >> 3, subR = t & 7;    // phase R: row, 8-col stripe
    const int colC = t >> 2, subC = t & 3;    // phase Cm: col, 8-row stripe
    const unsigned long long rm =
        ((const unsigned long long*)(refMask + rowBase))[subC];

    // running per-row top2 (dot products; larger dot == smaller distance)
    float bv0 = -INFINITY, bv1 = -INFINITY;
    int   bi0 = -1,        bi1 = -1;

    const int w    = t >> 5;        // wave id 0..7
    const int lane = t & 31;
    const int hl   = lane >> 4;     // half-wave select (K pair)
    const int l    = lane & 15;
    const int mB   = (w & 1) * 16;  // 2 row tiles
    const int nB   = (w >> 1) * 16; // 4 col tiles

    const float* ap = &As[(mB + l) * LDA + 2 * hl];
    const float* bp = &Bs[(nB + l) * LDA + 2 * hl];

    for (int tile0 = 0; tile0 < N; tile0 += BN) {
        stage_wait();
        __syncthreads();            // panels resident; prior-tile scans done

        // ---- 16x16 f32 WMMA tile per wave, K=256 in steps of 4 ----
        v8f c = {0.f, 0.f, 0.f, 0.f, 0.f, 0.f, 0.f, 0.f};
#pragma unroll 8
        for (int k = 0; k < KD; k += 4) {
            v2f a = *(const v2f*)(ap + k);
            v2f b = *(const v2f*)(bp + k);
            c = __builtin_amdgcn_wmma_f32_16x16x4_f32(
                    false, a, false, b, (short)0, c, false, false);
        }

        // Spill C: VGPR i -> row mB+i (lanes 0-15) / mB+i+8 (lanes 16-31), col nB+l
#pragma unroll
        for (int i = 0; i < 8; ++i)
            Cs[(mB + i + 8 * hl) * LDC + nB + l] = c[i];
        __syncthreads();            // all waves done reading Bs, Cs visible

        // ---- kick off next tile's B panel; DMA overlaps the scans below ----
        int next0 = tile0 + BN;
        if (next0 < N) {
            for (int q = t; q < BN * (KD / 4); q += 256) {
                int row = q / (KD / 4), c4 = q % (KD / 4);
                stage16(srcF + (size_t)(next0 + row) * KD + c4 * 4,
                        &Bs[row * LDA + c4 * 4]);
            }
        }

        // ---- per-row running top2 over this tile (branchless) ----
        {
            unsigned long long sm =
                ((const unsigned long long*)(srcMask + tile0))[subR];
            float4 va = *(const float4*)&Cs[rowR * LDC + subR * 8];
            float4 vb = *(const float4*)&Cs[rowR * LDC + subR * 8 + 4];
            float vals[8] = {va.x, va.y, va.z, va.w, vb.x, vb.y, vb.z, vb.w};
            int gbase = tile0 + subR * 8;
#pragma unroll
            for (int j = 0; j < 8; ++j) {
                bool ok  = ((sm >> (8 * j)) & 0xffull) != 0ull;
                float d  = vals[j];
                int   g  = gbase + j;
                bool gt0 = ok && (d > bv0);
                bool gt1 = ok && (d > bv1);
                float pv0 = bv0; int pi0 = bi0;
                bv0 = gt0 ? d : bv0;
                bi0 = gt0 ? g : bi0;
                bv1 = gt0 ? pv0 : (gt1 ? d : bv1);
                bi1 = gt0 ? pi0 : (gt1 ? g : bi1);
            }
        }
        // ---- per-column max over this block's rows (branchless) -> atomic ----
        {
            float best = -INFINITY; int bestr = -1;
#pragma unroll
            for (int j = 0; j < 8; ++j) {
                int r   = subC * 8 + j;
                float d = Cs[r * LDC + colC];
                bool ok = ((rm >> (8 * j)) & 0xffull) != 0ull;
                bool gt = ok && (d > best);
                best  = gt ? d : best;
                bestr = gt ? (rowBase + r) : bestr;
            }
            if (bestr >= 0) {
                unsigned long long p =
                    ((unsigned long long)f2ord(best) << 32) |
                    (unsigned)(~(unsigned)bestr);   // ~idx: ties pick lowest row
                atomicMax(&colBest[tile0 + colC], p);
            }
        }
    }

    // ---- merge the 8 per-row sub-stripe top2 candidates ----
    redV[(rowR * 8 + subR) * 2 + 0] = bv0;
    redV[(rowR * 8 + subR) * 2 + 1] = bv1;
    redI[(rowR * 8 + subR) * 2 + 0] = bi0;
    redI[(rowR * 8 + subR) * 2 + 1] = bi1;
    __syncthreads();
    if (t < BM) {
        float v0 = -INFINITY, v1 = -INFINITY; int i0 = -1, i1 = -1;
        for (int s = 0; s < 8; ++s) {
            for (int q = 0; q < 2; ++q) {
                float v = redV[(t * 8 + s) * 2 + q];
                int idx = redI[(t * 8 + s) * 2 + q];
                if (idx < 0) continue;
                if (v > v0 || (v == v0 && (unsigned)idx < (unsigned)i0)) {
                    v1 = v0; i1 = i0; v0 = v; i0 = idx;
                } else if (v > v1 || (v == v1 && (unsigned)idx < (unsigned)i1)) {
                    v1 = v; i1 = idx;
                }
            }
        }
        rowV0[rowBase + t] = v0;
        rowV1[rowBase + t] = v1;
        rowI0[rowBase + t] = i0;
    }
}

// ---------------------------------------------------------------------------
// per-row: ratio test + mutual NN + masked score, build sortable keys
// ---------------------------------------------------------------------------
__global__ void spm_post_kernel(
    const unsigned long long* __restrict__ colBest,
    const float* __restrict__ rowV0, const float* __restrict__ rowV1,
    const int* __restrict__ rowI0, const unsigned char* __restrict__ refMask,
    int M, unsigned long long* __restrict__ key, float* __restrict__ outScore)
{
    int m = blockIdx.x * 256 + threadIdx.x;
    if (m >= M) return;
    float v0 = rowV0[m], v1 = rowV1[m];
    int i0 = rowI0[m];
    float d0 = 2.0f - 2.0f * v0;
    float d1 = 2.0f - 2.0f * v1;
    bool validRatio = (d0 / (d1 + EPS)) < RATIO_THRESHOLD;
    bool mutual = false;
    if (i0 >= 0) {
        unsigned long long e = colBest[i0];
        unsigned mback = ~((unsigned)e);
        mutual = (mback == (unsigned)m);
    }
    bool valid = validRatio && mutual && (refMask[m] != 0) && (i0 >= 0);
    float ms = valid ? (1.0f / (d0 + EPS)) : -INFINITY;
    outScore[m] = ms;
    key[m] = ((unsigned long long)f2ord(ms) << 32) | (unsigned)(~(unsigned)m);
}

// ---------------------------------------------------------------------------
// exact sorted top-K by rank counting (keys unique: index packed in)
// ---------------------------------------------------------------------------
__global__ void spm_topk_kernel(
    const unsigned long long* __restrict__ key, const float* __restrict__ score,
    const int* __restrict__ rowI0, int M, int K, float* __restrict__ out)
{
    int m = blockIdx.x * 256 + threadIdx.x;
    if (m >= M) return;
    unsigned long long km = key[m];
    int rank = 0;
    for (int j = 0; j < M; ++j) rank += (key[j] > km) ? 1 : 0;
    if (rank < K) {
        out[rank]         = (float)m;              // ref_corr_indices
        out[K + rank]     = (float)rowI0[m];       // src_corr_indices
        out[2 * K + rank] = score[m];              // match_scores
    }
}

// ---------------------------------------------------------------------------
extern "C" void kernel_launch(void* const* d_in, const int* in_sizes, int n_in,
                              void* d_out, int out_size, void* d_ws, size_t ws_size,
                              hipStream_t stream) {
    const float* refF = (const float*)d_in[0];
    const float* srcF = (const float*)d_in[1];
    const unsigned char* refMask = (const unsigned char*)d_in[2];
    const unsigned char* srcMask = (const unsigned char*)d_in[3];
    const int M = in_sizes[2];
    const int N = in_sizes[3];
    const int K = out_size / 3;

    // workspace layout (8B-aligned base assumed from allocator)
    unsigned long long* colBest = (unsigned long long*)d_ws;
    float* rowV0 = (float*)(colBest + N);
    float* rowV1 = rowV0 + M;
    int*   rowI0 = (int*)(rowV1 + M);
    uintptr_t p = ((uintptr_t)(rowI0 + M) + 15u) & ~(uintptr_t)15u;
    unsigned long long* key = (unsigned long long*)p;
    float* outScore = (float*)(key + M);

    spm_init_kernel<<<(N + 255) / 256, 256, 0, stream>>>(colBest, N);
    spm_gemm_kernel<<<M / BM, 256, 0, stream>>>(refF, srcF, refMask, srcMask,
                                                M, N, colBest, rowV0, rowV1, rowI0);
    spm_post_kernel<<<(M + 255) / 256, 256, 0, stream>>>(colBest, rowV0, rowV1,
                                                         rowI0, refMask, M, key, outScore);
    spm_topk_kernel<<<(M + 255) / 256, 256, 0, stream>>>(key, outScore, rowI0,
                                                         M, K, (float*)d_out);
}